// GNNClassifier_412316860773
// MI455X (gfx1250) — compile-verified
//
#include <hip/hip_runtime.h>
#include <hip/hip_bf16.h>

#define VOCAB 30522
#define D 256
#define L 42
#define LP 48          // padded label count (3 x 16 N-tiles)
#define B 32
#define S 4096
#define W 4096

typedef __attribute__((ext_vector_type(2))) float v2f;
typedef __attribute__((ext_vector_type(8))) float v8f;

// ---------------------------------------------------------------------------
// Kernel 1: out[b,w,l] = cls_b[l]  (empty words must yield exactly cls_b);
//           counts[b,w] = 0
// ---------------------------------------------------------------------------
__global__ void init_kernel(float* __restrict__ out,
                            const float* __restrict__ cls_b,
                            int* __restrict__ counts) {
    const size_t stride = (size_t)gridDim.x * blockDim.x;
    size_t i = (size_t)blockIdx.x * blockDim.x + threadIdx.x;
    const size_t NOUT = (size_t)B * W * L;
    for (size_t idx = i; idx < NOUT; idx += stride)
        out[idx] = cls_b[idx % L];
    const size_t NCNT = (size_t)B * W;
    for (size_t idx = i; idx < NCNT; idx += stride)
        counts[idx] = 0;
}

// ---------------------------------------------------------------------------
// Kernel 2: counts[b, seg] += 1 for each wordpiece
// ---------------------------------------------------------------------------
__global__ void count_kernel(const int* __restrict__ seg_ids,
                             int* __restrict__ counts) {
    int i = blockIdx.x * blockDim.x + threadIdx.x;   // over B*S
    int b = i >> 12;                                  // i / S (S = 4096)
    atomicAdd(&counts[b * W + seg_ids[i]], 1);
}

// ---------------------------------------------------------------------------
// Kernel 3: fused gather + pos-add + WMMA GEMM (f32 16x16x4) + weighted
//           scatter-add into logits.
// Block = 256 threads = 8 waves; each wave: 16 pieces x 48 cols.
// ---------------------------------------------------------------------------
__global__ __launch_bounds__(256)
void gemm_scatter_kernel(const int*   __restrict__ input_ids,
                         const int*   __restrict__ seg_ids,
                         const float* __restrict__ emb_w,
                         const float* __restrict__ pos_w,
                         const float* __restrict__ cls_w,
                         const int*   __restrict__ counts,
                         float*       __restrict__ out) {
    __shared__ float lds[LP * D];   // cls_w transposed + zero-padded: lds[l*D + k]

    for (int e = threadIdx.x; e < LP * D; e += blockDim.x) {
        int l = e >> 8;             // e / D
        int k = e & (D - 1);        // e % D
        lds[e] = (l < L) ? cls_w[k * L + l] : 0.0f;
    }
    __syncthreads();

    const int wave = threadIdx.x >> 5;
    const int lane = threadIdx.x & 31;
    const int row  = lane & 15;     // A-matrix row M = lane & 15
    const int lh   = lane >> 4;     // lane half: selects K pair (0,1) vs (2,3)
    const int p0   = blockIdx.x * 128 + wave * 16;  // flat piece index of tile
    const int b    = p0 >> 12;      // p0 / S; a block never spans two batches

    // Per-lane A source pointers (gathered vocab row + position row)
    const int id   = input_ids[p0 + row];
    const int sidx = (p0 + row) & (S - 1);          // position within sequence
    const v2f* ea = (const v2f*)(emb_w + (size_t)id   * D) + lh;
    const v2f* pa = (const v2f*)(pos_w + (size_t)sidx * D) + lh;
    const v2f* lb = (const v2f*)lds;

    const int n = lane & 15;        // output column within a 16-wide N tile

    v8f acc0 = {}, acc1 = {}, acc2 = {};

#pragma unroll 8
    for (int j = 0; j < 64; ++j) {  // K = 256, 4 per WMMA step
        v2f a  = ea[2 * j] + pa[2 * j];
        v2f b0 = lb[(n +  0) * 128 + 2 * j + lh];
        v2f b1 = lb[(n + 16) * 128 + 2 * j + lh];
        v2f b2 = lb[(n + 32) * 128 + 2 * j + lh];
        acc0 = __builtin_amdgcn_wmma_f32_16x16x4_f32(false, a, false, b0,
                                                     (short)0, acc0, false, false);
        acc1 = __builtin_amdgcn_wmma_f32_16x16x4_f32(false, a, false, b1,
                                                     (short)0, acc1, false, false);
        acc2 = __builtin_amdgcn_wmma_f32_16x16x4_f32(false, a, false, b2,
                                                     (short)0, acc2, false, false);
    }

    // D layout: VGPR r holds row M = r + 8*lh, column N = lane & 15.
    // Scale each piece's logits by 1/count and scatter-add into its word.
#pragma unroll
    for (int r = 0; r < 8; ++r) {
        int   m    = r + 8 * lh;
        int   seg  = seg_ids[p0 + m];
        float inv  = 1.0f / (float)counts[b * W + seg];
        float* ob  = out + ((size_t)b * W + seg) * L;
        atomicAdd(ob + n,      acc0[r] * inv);      // cols  0..15  (< 42)
        atomicAdd(ob + n + 16, acc1[r] * inv);      // cols 16..31  (< 42)
        if (n + 32 < L)                             // cols 32..47: keep < 42
            atomicAdd(ob + n + 32, acc2[r] * inv);
    }
}

// ---------------------------------------------------------------------------
extern "C" void kernel_launch(void* const* d_in, const int* in_sizes, int n_in,
                              void* d_out, int out_size, void* d_ws, size_t ws_size,
                              hipStream_t stream) {
    const int*   input_ids = (const int*)  d_in[0];   // [B,S]
    const int*   seg_ids   = (const int*)  d_in[1];   // [B,S]
    const float* emb_w     = (const float*)d_in[2];   // [VOCAB,D]
    const float* pos_w     = (const float*)d_in[3];   // [S,D]
    const float* cls_w     = (const float*)d_in[4];   // [D,L]
    const float* cls_b     = (const float*)d_in[5];   // [L]
    float*       out       = (float*)d_out;           // [B,W,L]
    int*         counts    = (int*)d_ws;               // [B,W] = 512 KB scratch

    (void)in_sizes; (void)n_in; (void)out_size; (void)ws_size;

    // 1) seed output with bias, zero counts
    init_kernel<<<2048, 256, 0, stream>>>(out, cls_b, counts);
    // 2) per-word piece counts
    count_kernel<<<(B * S) / 256, 256, 0, stream>>>(seg_ids, counts);
    // 3) fused WMMA GEMM + weighted scatter
    gemm_scatter_kernel<<<(B * S) / 128, 256, 0, stream>>>(
        input_ids, seg_ids, emb_w, pos_w, cls_w, counts, out);
}